// Detection_50869592654784
// MI455X (gfx1250) — compile-verified
//
#include <hip/hip_runtime.h>
#include <hip/hip_bf16.h>

#define NPRIORS      4000000
#define TOPK         200
#define KEEPK        100
#define CONF_TH      0.01f
#define NMS_TH       0.45f
#define NBINS        16384
#define CAP          4096
#define HIST_BLOCKS  1024
#define HIST_THREADS 256

// ---- CDNA5 async global->LDS DMA (ASYNCcnt path), inline asm per ISA 15.18 ----
// Pass-1 variant: TH_LOAD_HT keeps conf (32 MB) resident in the 192 MB L2 so the
// second streaming pass hits L2 instead of HBM.
__device__ __forceinline__ void async_load_f4_ht(unsigned lds_addr, const void* gaddr) {
  asm volatile("global_load_async_to_lds_b128 %0, %1, off th:TH_LOAD_HT"
               :: "v"(lds_addr), "v"((unsigned long long)(size_t)gaddr)
               : "memory");
}
__device__ __forceinline__ void async_load_f4(unsigned lds_addr, const void* gaddr) {
  asm volatile("global_load_async_to_lds_b128 %0, %1, off"
               :: "v"(lds_addr), "v"((unsigned long long)(size_t)gaddr)
               : "memory");
}
__device__ __forceinline__ void wait_async0() { asm volatile("s_wait_asynccnt 0x0" ::: "memory"); }
__device__ __forceinline__ void wait_async1() { asm volatile("s_wait_asynccnt 0x1" ::: "memory"); }

__device__ __forceinline__ int bin_of(float s) {
  int b = (int)(s * (float)NBINS);
  b = b < 0 ? 0 : (b > NBINS - 1 ? NBINS - 1 : b);
  return b;
}

// ws layout: hist[NBINS] | counter | cutoff | cand_score[CAP] | cand_idx[CAP]
__global__ void init_kernel(int* hist) {
  int i = blockIdx.x * blockDim.x + threadIdx.x;
  int stride = gridDim.x * blockDim.x;
  for (; i < NBINS + 2; i += stride) hist[i] = 0;
}

// Pass 1: histogram of class-1 scores above threshold.
// Double-buffered async DMA of conf tiles (float4 = 2 priors/lane) into LDS.
__global__ void __launch_bounds__(HIST_THREADS)
hist_kernel(const float* __restrict__ conf, int* __restrict__ hist) {
  __shared__ float4 buf[2][HIST_THREADS];
  const int nPairs = NPRIORS / 2;
  const int nTiles = (nPairs + HIST_THREADS - 1) / HIST_THREADS;
  const float4* conf4 = (const float4*)conf;
  const int tid = threadIdx.x;
  int tile = blockIdx.x;
  int pb = 0;
  if (tile < nTiles) {
    int p = tile * HIST_THREADS + tid;
    if (p > nPairs - 1) p = nPairs - 1;
    async_load_f4_ht((unsigned)(size_t)&buf[0][tid], conf4 + p);
  }
  for (; tile < nTiles; tile += gridDim.x) {
    int next = tile + gridDim.x;
    if (next < nTiles) {
      int p = next * HIST_THREADS + tid;
      if (p > nPairs - 1) p = nPairs - 1;
      async_load_f4_ht((unsigned)(size_t)&buf[pb ^ 1][tid], conf4 + p);
      wait_async1();            // oldest (current tile) complete
    } else {
      wait_async0();
    }
    float4 v = buf[pb][tid];    // v.y = score(2p), v.w = score(2p+1)
    int p = tile * HIST_THREADS + tid;
    if (p < nPairs) {
      if (v.y > CONF_TH) atomicAdd(&hist[bin_of(v.y)], 1);
      if (v.w > CONF_TH) atomicAdd(&hist[bin_of(v.w)], 1);
    }
    pb ^= 1;
  }
}

// Pass 2: largest bin b with suffix-count >= TOPK  => top-200 fully contained in bins >= b.
__global__ void cutoff_kernel(const int* __restrict__ hist, int* __restrict__ cutoff) {
  __shared__ int csum[256];
  const int t = threadIdx.x;
  const int per = NBINS / 256;
  int s = 0;
  for (int b = t * per; b < t * per + per; ++b) s += hist[b];
  csum[t] = s;
  __syncthreads();
  if (t == 0) {
    int acc = 0, cb = 0;
    bool found = false;
    for (int c = 255; c >= 0 && !found; --c) {
      if (acc + csum[c] >= TOPK) {
        int a2 = acc;
        for (int b = c * per + per - 1; b >= c * per; --b) {
          a2 += hist[b];
          if (a2 >= TOPK) { cb = b; found = true; break; }
        }
        if (!found) { cb = c * per; found = true; }
      } else {
        acc += csum[c];
      }
    }
    *cutoff = found ? cb : 0;
  }
}

// Pass 3: collect (score, prior index) for bins >= cutoff. conf should hit L2 now.
__global__ void __launch_bounds__(HIST_THREADS)
collect_kernel(const float* __restrict__ conf, const int* __restrict__ cutoff,
               int* __restrict__ counter, float* __restrict__ cand_s,
               int* __restrict__ cand_i) {
  __shared__ float4 buf[2][HIST_THREADS];
  const int cb = *cutoff;
  const int nPairs = NPRIORS / 2;
  const int nTiles = (nPairs + HIST_THREADS - 1) / HIST_THREADS;
  const float4* conf4 = (const float4*)conf;
  const int tid = threadIdx.x;
  int tile = blockIdx.x;
  int pb = 0;
  if (tile < nTiles) {
    int p = tile * HIST_THREADS + tid;
    if (p > nPairs - 1) p = nPairs - 1;
    async_load_f4((unsigned)(size_t)&buf[0][tid], conf4 + p);
  }
  for (; tile < nTiles; tile += gridDim.x) {
    int next = tile + gridDim.x;
    if (next < nTiles) {
      int p = next * HIST_THREADS + tid;
      if (p > nPairs - 1) p = nPairs - 1;
      async_load_f4((unsigned)(size_t)&buf[pb ^ 1][tid], conf4 + p);
      wait_async1();
    } else {
      wait_async0();
    }
    float4 v = buf[pb][tid];
    int p = tile * HIST_THREADS + tid;
    if (p < nPairs) {
      if (v.y > CONF_TH && bin_of(v.y) >= cb) {
        int pos = atomicAdd(counter, 1);
        if (pos < CAP) { cand_s[pos] = v.y; cand_i[pos] = 2 * p; }
      }
      if (v.w > CONF_TH && bin_of(v.w) >= cb) {
        int pos = atomicAdd(counter, 1);
        if (pos < CAP) { cand_s[pos] = v.w; cand_i[pos] = 2 * p + 1; }
      }
    }
    pb ^= 1;
  }
}

// Pass 4: sort candidates (desc score, asc idx), decode top-200, greedy NMS, emit 100x7.
__global__ void __launch_bounds__(1024)
finalize_kernel(const float* __restrict__ loc, const float* __restrict__ prior,
                const int* __restrict__ counter, const float* __restrict__ cand_s,
                const int* __restrict__ cand_i, float* __restrict__ out) {
  __shared__ unsigned long long s_key[CAP];
  __shared__ float sx1[TOPK], sy1[TOPK], sx2[TOPK], sy2[TOPK], sar[TOPK], ssc[TOPK];
  __shared__ int skeep[TOPK];
  __shared__ int s_ord[KEEPK];
  __shared__ int s_num;
  const int tid = threadIdx.x;

  int count = *counter;
  if (count > CAP) count = CAP;
  for (int m = tid; m < CAP; m += blockDim.x) {
    unsigned long long key = 0ull;
    if (m < count) {
      unsigned fb = __float_as_uint(cand_s[m]);     // scores > 0 => bit order == value order
      key = ((unsigned long long)fb << 32) |
            (unsigned long long)(0xFFFFFFFFu - (unsigned)cand_i[m]);  // lower idx wins ties
    }
    s_key[m] = key;
  }
  // Bitonic sort, descending.
  for (int k = 2; k <= CAP; k <<= 1) {
    for (int j = k >> 1; j > 0; j >>= 1) {
      __syncthreads();
      for (int i = tid; i < CAP; i += blockDim.x) {
        int ixj = i ^ j;
        if (ixj > i) {
          unsigned long long a = s_key[i], b = s_key[ixj];
          bool up = ((i & k) == 0);
          if (up ? (a < b) : (a > b)) { s_key[i] = b; s_key[ixj] = a; }
        }
      }
    }
  }
  __syncthreads();

  if (tid < TOPK) {
    unsigned long long key = s_key[tid];
    int valid = (key != 0ull);
    float sc = 0.f, x1 = 0.f, y1 = 0.f, x2 = 0.f, y2 = 0.f;
    if (valid) {
      sc = __uint_as_float((unsigned)(key >> 32));
      int idx = (int)(0xFFFFFFFFu - (unsigned)(key & 0xFFFFFFFFull));
      float4 l  = ((const float4*)loc)[idx];    // (lx, ly, lw, lh)
      float4 pr = ((const float4*)prior)[idx];  // (xmin, ymin, xmax, ymax)
      float cx = (pr.z + pr.x) * 0.5f;
      float cy = (pr.w + pr.y) * 0.5f;
      float pw = pr.z - pr.x;
      float ph = pr.w - pr.y;
      float ncx = cx + l.x * 0.1f * pw;
      float ncy = cy + l.y * 0.1f * ph;
      float nw = pw * expf(l.z * 0.2f);
      float nh = ph * expf(l.w * 0.2f);
      x1 = ncx - nw * 0.5f; y1 = ncy - nh * 0.5f;
      x2 = x1 + nw;         y2 = y1 + nh;
    }
    sx1[tid] = x1; sy1[tid] = y1; sx2[tid] = x2; sy2[tid] = y2;
    sar[tid] = (x2 - x1) * (y2 - y1);
    ssc[tid] = sc;
    skeep[tid] = valid;
  }
  // Greedy NMS: iteration i only writes indices > i; barrier at loop head orders reads of slot i.
  for (int i = 0; i < TOPK; ++i) {
    __syncthreads();
    if (tid < TOPK && tid > i && skeep[i]) {
      float xx1 = fmaxf(sx1[tid], sx1[i]);
      float yy1 = fmaxf(sy1[tid], sy1[i]);
      float xx2 = fminf(sx2[tid], sx2[i]);
      float yy2 = fminf(sy2[tid], sy2[i]);
      float inter = fmaxf(xx2 - xx1, 0.f) * fmaxf(yy2 - yy1, 0.f);
      float uni = sar[tid] - inter + sar[i];
      if (inter / uni > NMS_TH) skeep[tid] = 0;
    }
  }
  __syncthreads();
  if (tid == 0) {   // stable compaction == argsort(rank)[:100]
    int c = 0;
    for (int i = 0; i < TOPK && c < KEEPK; ++i)
      if (skeep[i]) s_ord[c++] = i;
    s_num = c;
  }
  __syncthreads();
  if (tid < KEEPK) {
    float r1 = 0.f, r2 = 0.f, r3 = 0.f, r4 = 0.f, r5 = 0.f, r6 = 0.f;
    if (tid < s_num) {
      int i = s_ord[tid];
      r1 = 1.0f; r2 = ssc[i];
      r3 = sx1[i]; r4 = sy1[i]; r5 = sx2[i]; r6 = sy2[i];
    }
    float* o = out + tid * 7;
    o[0] = 0.f; o[1] = r1; o[2] = r2; o[3] = r3; o[4] = r4; o[5] = r5; o[6] = r6;
  }
}

extern "C" void kernel_launch(void* const* d_in, const int* in_sizes, int n_in,
                              void* d_out, int out_size, void* d_ws, size_t ws_size,
                              hipStream_t stream) {
  (void)in_sizes; (void)n_in; (void)out_size; (void)ws_size;
  const float* loc   = (const float*)d_in[0];
  const float* conf  = (const float*)d_in[1];
  const float* prior = (const float*)d_in[2];
  float* out = (float*)d_out;

  int*   hist    = (int*)d_ws;
  int*   counter = hist + NBINS;
  int*   cutoff  = hist + NBINS + 1;
  float* cand_s  = (float*)(hist + NBINS + 2);
  int*   cand_i  = (int*)(cand_s + CAP);

  init_kernel<<<64, 256, 0, stream>>>(hist);
  hist_kernel<<<HIST_BLOCKS, HIST_THREADS, 0, stream>>>(conf, hist);
  cutoff_kernel<<<1, 256, 0, stream>>>(hist, cutoff);
  collect_kernel<<<HIST_BLOCKS, HIST_THREADS, 0, stream>>>(conf, cutoff, counter, cand_s, cand_i);
  finalize_kernel<<<1, 1024, 0, stream>>>(loc, prior, counter, cand_s, cand_i, out);
}